// CustomLoss_45449343926664
// MI455X (gfx1250) — compile-verified
//
#include <hip/hip_runtime.h>

typedef __attribute__((ext_vector_type(2))) float v2f;
typedef __attribute__((ext_vector_type(8))) float v8f;

#define N_ROWS 8192
#define N_DIM  128
#define CHUNK_ROWS 64
#define LDS_STRIDE 132   // 128 + 4 pad dwords -> 528B row stride: 16B-aligned, conflict-free b64 frag loads

// ---------------- kernel 1: per-row sum of squares for both clusters ----------------
__global__ void row_sumsq_kernel(const float* __restrict__ c1,
                                 const float* __restrict__ c2,
                                 float* __restrict__ sq1,
                                 float* __restrict__ sq2) {
  const int wave = threadIdx.x >> 5;
  const int lane = threadIdx.x & 31;
  const int sel  = blockIdx.x >> 10;              // 0 -> c1, 1 -> c2
  const int row  = (blockIdx.x & 1023) * 8 + wave;
  const float* src = sel ? c2 : c1;
  const float4 v = *(const float4*)(src + (size_t)row * N_DIM + lane * 4);
  float s = v.x * v.x + v.y * v.y + v.z * v.z + v.w * v.w;
#pragma unroll
  for (int m = 16; m >= 1; m >>= 1) s += __shfl_xor(s, m, 32);
  if (lane == 0) (sel ? sq2 : sq1)[row] = s;
}

// ---------------- kernel 2: per-column partial sums (deterministic, no atomics) ------
__global__ void col_stats_kernel(const float* __restrict__ c1,
                                 const float* __restrict__ c2,
                                 float* __restrict__ ps1,
                                 float* __restrict__ ps2,
                                 float* __restrict__ pq1) {
  const int k = threadIdx.x;                      // 0..127 (feature)
  const int b = blockIdx.x;                       // 0..63 (row chunk of 128)
  const float* p1 = c1 + (size_t)b * 128 * N_DIM + k;
  const float* p2 = c2 + (size_t)b * 128 * N_DIM + k;
  float s1 = 0.f, s2 = 0.f, q1 = 0.f;
  for (int r = 0; r < 128; ++r) {
    float a  = p1[r * N_DIM];
    float bb = p2[r * N_DIM];
    s1 += a; q1 += a * a; s2 += bb;
  }
  ps1[b * N_DIM + k] = s1;
  ps2[b * N_DIM + k] = s2;
  pq1[b * N_DIM + k] = q1;
}

// ---------------- kernel 3: fused WMMA cross-product + min-distance ------------------
// 64 blocks x 256 threads (8 waves). Each wave: 16-row c1 tile (A frags in regs),
// streams all of c2 through LDS, 4 concurrent 16x16 WMMA accumulator chains.
__global__ __launch_bounds__(256)
void min_dist_kernel(const float* __restrict__ c1,
                     const float* __restrict__ c2,
                     const float* __restrict__ sq1,
                     const float* __restrict__ sq2,
                     float* __restrict__ pdist) {
  __shared__ float lds[CHUNK_ROWS * LDS_STRIDE];
  const int tid  = threadIdx.x;
  const int wave = tid >> 5;
  const int lane = tid & 31;
  const int n    = lane & 15;       // M index for A, N index for B/C
  const int hi   = lane >> 4;       // K-pair selector per ISA layout
  const int rowBase = blockIdx.x * 128 + wave * 16;

  // A fragments: 16x128 c1 tile, ISA layout: lane (n,hi) holds A[n][4kk+2hi .. +1]
  v2f afr[32];
  {
    const float* ap = c1 + (size_t)(rowBase + n) * N_DIM + 2 * hi;
#pragma unroll
    for (int kk = 0; kk < 32; ++kk)
      afr[kk] = *(const v2f*)(ap + kk * 4);
  }

  float s1v[8];
#pragma unroll
  for (int v = 0; v < 8; ++v) s1v[v] = sq1[rowBase + v + 8 * hi];

  float minv[8];
#pragma unroll
  for (int v = 0; v < 8; ++v) minv[v] = 3.4e38f;

  const v8f vzero = {0.f, 0.f, 0.f, 0.f, 0.f, 0.f, 0.f, 0.f};

  for (int jc = 0; jc < N_ROWS / CHUNK_ROWS; ++jc) {
    // cooperative, coalesced load of a 64-row c2 chunk into padded LDS
    {
      const float* src = c2 + (size_t)jc * CHUNK_ROWS * N_DIM;
#pragma unroll
      for (int i = 0; i < 8; ++i) {
        int f   = (i * 256 + tid) * 4;
        int row = f >> 7;
        int col = f & 127;
        float4 v = *(const float4*)(src + row * N_DIM + col);
        *(float4*)(lds + row * LDS_STRIDE + col) = v;   // 528B row stride keeps 16B align
      }
    }
    __syncthreads();

    v8f acc[4];
#pragma unroll
    for (int t = 0; t < 4; ++t) acc[t] = vzero;

    const v2f* bp[4];
#pragma unroll
    for (int t = 0; t < 4; ++t)
      bp[t] = (const v2f*)(lds + (t * 16 + n) * LDS_STRIDE) + hi;

    // K loop: 32 x (4 independent WMMA chains) — hides fp32 WMMA latency
#pragma unroll
    for (int kk = 0; kk < 32; ++kk) {
#pragma unroll
      for (int t = 0; t < 4; ++t) {
        v2f b = bp[t][kk * 2];
        acc[t] = __builtin_amdgcn_wmma_f32_16x16x4_f32(
            false, afr[kk], false, b, (short)0, acc[t], false, false);
      }
    }

    // fused epilogue: d2 = ||a||^2 - 2*cross + ||b||^2, running min
#pragma unroll
    for (int t = 0; t < 4; ++t) {
      float s2v = sq2[jc * CHUNK_ROWS + t * 16 + n];
#pragma unroll
      for (int v = 0; v < 8; ++v) {
        float d2 = s1v[v] - 2.f * acc[t][v] + s2v;
        minv[v] = fminf(minv[v], d2);
      }
    }
    __syncthreads();
  }

  // C layout: VGPR v holds row M=v (lanes 0-15) / M=v+8 (lanes 16-31).
  // Min-reduce each 16-lane half, then sum 16 row-mins per wave.
#pragma unroll
  for (int v = 0; v < 8; ++v) {
    float x = minv[v];
#pragma unroll
    for (int m = 8; m >= 1; m >>= 1) x = fminf(x, __shfl_xor(x, m, 32));
    minv[v] = x;
  }
  float tsum = 0.f;
#pragma unroll
  for (int v = 0; v < 8; ++v) tsum += minv[v];
  tsum += __shfl_xor(tsum, 16, 32);
  if (lane == 0) pdist[blockIdx.x * 8 + wave] = tsum;
}

// ---------------- kernel 4: deterministic final combine ------------------------------
__global__ void finalize_kernel(const float* __restrict__ ps1,
                                const float* __restrict__ ps2,
                                const float* __restrict__ pq1,
                                const float* __restrict__ pdist,
                                float* __restrict__ out) {
  __shared__ float red[128];
  const int k = threadIdx.x;   // 0..127
  float s1 = 0.f, s2 = 0.f, q1 = 0.f;
  for (int b = 0; b < 64; ++b) {
    s1 += ps1[b * 128 + k];
    s2 += ps2[b * 128 + k];
    q1 += pq1[b * 128 + k];
  }
  const float invN = 1.f / 8192.f;
  float m1  = s1 * invN, m2 = s2 * invN;
  float dm  = (m1 - m2) * (m1 - m2);          // mean-loss contribution
  float var = q1 * invN - m1 * m1;
  float dsp = fmaxf(0.1f - var, 0.f);         // dispersion contribution
  float dst = 0.f;
  for (int i = k; i < 512; i += 128) dst += pdist[i];

  red[k] = dm * (1.f / 128.f) + dsp * (1.f / 128.f) + dst * invN;
  __syncthreads();
  for (int s = 64; s >= 1; s >>= 1) {
    if (k < s) red[k] += red[k + s];
    __syncthreads();
  }
  if (k == 0) out[0] = red[0];
}

extern "C" void kernel_launch(void* const* d_in, const int* in_sizes, int n_in,
                              void* d_out, int out_size, void* d_ws, size_t ws_size,
                              hipStream_t stream) {
  const float* c1 = (const float*)d_in[0];
  const float* c2 = (const float*)d_in[1];
  float* ws    = (float*)d_ws;
  float* sq1   = ws;                 // 8192
  float* sq2   = ws + 8192;          // 8192
  float* ps1   = ws + 16384;         // 64*128
  float* ps2   = ps1 + 8192;         // 64*128
  float* pq1   = ps2 + 8192;         // 64*128
  float* pdist = pq1 + 8192;         // 512

  row_sumsq_kernel<<<2048, 256, 0, stream>>>(c1, c2, sq1, sq2);
  col_stats_kernel<<<64, 128, 0, stream>>>(c1, c2, ps1, ps2, pq1);
  min_dist_kernel<<<64, 256, 0, stream>>>(c1, c2, sq1, sq2, pdist);
  finalize_kernel<<<1, 128, 0, stream>>>(ps1, ps2, pq1, pdist, (float*)d_out);
}